// house_classifier_90185723282019
// MI455X (gfx1250) — compile-verified
//
#include <hip/hip_runtime.h>
#include <math.h>

// ---------------- problem constants (from reference) ----------------
#define N_NODES   100000
#define N_EDGES   600000
#define N_GRAPHS  512
#define DIM       128
#define NODE_TILES (N_NODES / 16)   // 6250, exact
#define LDS_STRIDE 144              // 128 + 16: kills 2-row bank aliasing between lane halves

typedef __attribute__((ext_vector_type(2))) float v2f;
typedef __attribute__((ext_vector_type(8))) float v8f;

// ---------------- utility: zero a float4 region ----------------
__global__ void zero_f4_kernel(float4* __restrict__ p, int n4) {
    int i = blockIdx.x * blockDim.x + threadIdx.x;
    if (i < n4) p[i] = make_float4(0.f, 0.f, 0.f, 0.f);
}

// ---------------- in-degree counts (shared by all 3 layers) ----------------
__global__ void count_edges_kernel(const int* __restrict__ dst, float* __restrict__ cnt) {
    int e = blockIdx.x * blockDim.x + threadIdx.x;
    if (e < N_EDGES) atomicAdd(&cnt[dst[e]], 1.0f);
}

// ---------------- scatter-add h[src] into agg[dst] ----------------
// one thread per (edge, 4-feature chunk): coalesced float4 gather + 4 float atomics
__global__ void scatter_add_kernel(const float* __restrict__ h,
                                   const int* __restrict__ src,
                                   const int* __restrict__ dst,
                                   float* __restrict__ agg) {
    int t = blockIdx.x * blockDim.x + threadIdx.x;       // < E*32 = 19.2M
    int e  = t >> 5;
    int fc = t & 31;
    if (e >= N_EDGES) return;
    int s = src[e], d = dst[e];
    float4 v = *(const float4*)(h + (size_t)s * DIM + fc * 4);
    float* o = agg + (size_t)d * DIM + fc * 4;
    atomicAdd(o + 0, v.x);
    atomicAdd(o + 1, v.y);
    atomicAdd(o + 2, v.z);
    atomicAdd(o + 3, v.w);
}

// ---------------- fused SAGE layer: relu(mean@Wl + h@Wr + b) via f32 WMMA ----------------
// One wave per 16-node tile; 8 waves/block. Weights staged in LDS.
__launch_bounds__(256)
__global__ void sage_gemm_kernel(const float* __restrict__ h_in,
                                 const float* __restrict__ agg,
                                 const float* __restrict__ cnt,
                                 const float* __restrict__ Wl,
                                 const float* __restrict__ Wr,
                                 const float* __restrict__ bias,
                                 float* __restrict__ h_out) {
    __shared__ float sWl[DIM * LDS_STRIDE];
    __shared__ float sWr[DIM * LDS_STRIDE];
    __shared__ float sB[DIM];

    const int tid = threadIdx.x;
    // cooperative load of weights into LDS (float4, padded row stride)
    for (int i = tid; i < DIM * (DIM / 4); i += 256) {   // 4096 float4s per matrix
        int k = i >> 5;          // row 0..127
        int c = i & 31;          // float4 within row
        ((float4*)(sWl + k * LDS_STRIDE))[c] = ((const float4*)(Wl + (size_t)k * DIM))[c];
        ((float4*)(sWr + k * LDS_STRIDE))[c] = ((const float4*)(Wr + (size_t)k * DIM))[c];
    }
    if (tid < DIM) sB[tid] = bias[tid];
    __syncthreads();

    const int wave = tid >> 5;
    const int lane = tid & 31;
    const int tile = blockIdx.x * 8 + wave;
    if (tile >= NODE_TILES) return;   // wave-uniform branch; EXEC stays all-ones for WMMA

    const int m  = lane & 15;   // A-fragment row / B&C column within tile
    const int kh = lane >> 4;   // K half (lanes 16-31 carry K = +2)
    const int node = tile * 16 + m;

    const float inv = 1.0f / fmaxf(cnt[node], 1.0f);

    // Preload A fragments for both GEMMs: ISA 16x4 f32 A layout,
    // lane half kh holds {K=4*ks+2*kh, K=4*ks+2*kh+1} of row m.
    v2f a_h[32], a_m[32];
    const float* hrow = h_in + (size_t)node * DIM;
    const float* arow = agg  + (size_t)node * DIM;
#pragma unroll
    for (int ks = 0; ks < 32; ++ks) {
        const int off = ks * 4 + kh * 2;
        a_h[ks] = *(const v2f*)(hrow + off);
        v2f am  = *(const v2f*)(arow + off);
        am.x *= inv; am.y *= inv;
        a_m[ks] = am;
    }

    for (int nt = 0; nt < 8; ++nt) {            // 8 column tiles of 16
        const int col = nt * 16 + m;
        v8f acc = {0.f, 0.f, 0.f, 0.f, 0.f, 0.f, 0.f, 0.f};
#pragma unroll
        for (int ks = 0; ks < 32; ++ks) {       // K = 128 in steps of 4
            const int k0 = ks * 4 + kh * 2;
            v2f bl, br;                          // B 4x16 frag (mirrors A layout)
            bl.x = sWl[(k0    ) * LDS_STRIDE + col];
            bl.y = sWl[(k0 + 1) * LDS_STRIDE + col];
            br.x = sWr[(k0    ) * LDS_STRIDE + col];
            br.y = sWr[(k0 + 1) * LDS_STRIDE + col];
            acc = __builtin_amdgcn_wmma_f32_16x16x4_f32(
                false, a_m[ks], false, bl, (short)0, acc, false, false);
            acc = __builtin_amdgcn_wmma_f32_16x16x4_f32(
                false, a_h[ks], false, br, (short)0, acc, false, false);
        }
        const float bv = sB[col];
#pragma unroll
        for (int r = 0; r < 8; ++r) {
            // C/D layout: VGPR r, lanes 0-15 -> M=r, lanes 16-31 -> M=r+8; N = lane&15
            const int row = tile * 16 + r + kh * 8;
            float v = acc[r] + bv;
            h_out[(size_t)row * DIM + col] = v > 0.f ? v : 0.f;
        }
    }
}

// ---------------- per-graph sum pooling via atomics ----------------
__global__ void pool_kernel(const float* __restrict__ h,
                            const int* __restrict__ batch,
                            float* __restrict__ pooled) {
    int t = blockIdx.x * blockDim.x + threadIdx.x;   // < N*32
    int node = t >> 5;
    int fc = t & 31;
    if (node >= N_NODES) return;
    int g = batch[node];
    float4 v = *(const float4*)(h + (size_t)node * DIM + fc * 4);
    float* o = pooled + (size_t)g * DIM + fc * 4;
    atomicAdd(o + 0, v.x);
    atomicAdd(o + 1, v.y);
    atomicAdd(o + 2, v.z);
    atomicAdd(o + 3, v.w);
}

// ---------------- readout head: sigmoid(pooled @ Wro + bro) ----------------
__global__ void head_kernel(const float* __restrict__ pooled,
                            const float* __restrict__ Wro,
                            const float* __restrict__ bro,
                            float* __restrict__ out) {
    int wave = threadIdx.x >> 5;
    int lane = threadIdx.x & 31;
    int g = blockIdx.x * 8 + wave;
    if (g >= N_GRAPHS) return;
    const float* pr = pooled + (size_t)g * DIM;
    float s = 0.f;
#pragma unroll
    for (int i = 0; i < 4; ++i) {
        int f = lane * 4 + i;
        s += pr[f] * Wro[f];
    }
#pragma unroll
    for (int off = 16; off > 0; off >>= 1) s += __shfl_xor(s, off, 32);
    if (lane == 0) out[g] = 1.0f / (1.0f + expf(-(s + bro[0])));
}

// ---------------- host launcher ----------------
extern "C" void kernel_launch(void* const* d_in, const int* in_sizes, int n_in,
                              void* d_out, int out_size, void* d_ws, size_t ws_size,
                              hipStream_t stream) {
    (void)in_sizes; (void)n_in; (void)out_size; (void)ws_size;

    const float* x     = (const float*)d_in[0];
    const int*   ei    = (const int*)d_in[1];
    const int*   batch = (const int*)d_in[2];
    const float* Wl[3] = {(const float*)d_in[3], (const float*)d_in[6], (const float*)d_in[9]};
    const float* Wr[3] = {(const float*)d_in[4], (const float*)d_in[7], (const float*)d_in[10]};
    const float* bb[3] = {(const float*)d_in[5], (const float*)d_in[8], (const float*)d_in[11]};
    const float* Wro   = (const float*)d_in[12];
    const float* bro   = (const float*)d_in[13];
    float* out = (float*)d_out;

    // workspace layout (floats): agg | hA | hB | cnt | pooled
    float* agg    = (float*)d_ws;
    float* hA     = agg + (size_t)N_NODES * DIM;
    float* hB     = hA  + (size_t)N_NODES * DIM;
    float* cnt    = hB  + (size_t)N_NODES * DIM;
    float* pooled = cnt + N_NODES;

    const int* src = ei;
    const int* dst = ei + N_EDGES;

    // zero cnt + pooled (contiguous: 100000 + 65536 floats, both %4==0)
    {
        int n4 = (N_NODES + N_GRAPHS * DIM) / 4;
        zero_f4_kernel<<<(n4 + 255) / 256, 256, 0, stream>>>((float4*)cnt, n4);
    }
    count_edges_kernel<<<(N_EDGES + 255) / 256, 256, 0, stream>>>(dst, cnt);

    const float* hin = x;
    float* hout = hA;
    for (int L = 0; L < 3; ++L) {
        int n4 = N_NODES * DIM / 4;  // 3.2M
        zero_f4_kernel<<<(n4 + 255) / 256, 256, 0, stream>>>((float4*)agg, n4);
        scatter_add_kernel<<<(N_EDGES * 32) / 256, 256, 0, stream>>>(hin, src, dst, agg);
        sage_gemm_kernel<<<(NODE_TILES + 7) / 8, 256, 0, stream>>>(
            hin, agg, cnt, Wl[L], Wr[L], bb[L], hout);
        hin = hout;
        hout = (L == 0) ? hB : hA;
    }
    // hin now points at hA (x->hA->hB->hA)
    pool_kernel<<<(N_NODES * 32) / 256, 256, 0, stream>>>(hin, batch, pooled);
    head_kernel<<<(N_GRAPHS + 7) / 8, 256, 0, stream>>>(pooled, Wro, bro, out);
}